// MultiModalBreastCancerModel_16346645529112
// MI455X (gfx1250) — compile-verified
//
#include <hip/hip_runtime.h>
#include <math.h>

typedef __attribute__((ext_vector_type(2))) float v2f;
typedef __attribute__((ext_vector_type(8))) float v8f;

#define L_SIG 16384
#define B_ROWS 4096
#define FT_THREADS 1024

__device__ __forceinline__ float wave_sum(float v) {
    #pragma unroll
    for (int o = 16; o > 0; o >>= 1) v += __shfl_down(v, o, 32);
    return v;
}
__device__ __forceinline__ float wave_max(float v) {
    #pragma unroll
    for (int o = 16; o > 0; o >>= 1) v = fmaxf(v, __shfl_down(v, o, 32));
    return v;
}

__device__ float block_sum(float v, float* buf, int tid) {
    __syncthreads();
    v = wave_sum(v);
    if ((tid & 31) == 0) buf[tid >> 5] = v;
    __syncthreads();
    if (tid == 0) {
        float t = 0.f;
        #pragma unroll
        for (int w = 0; w < FT_THREADS / 32; ++w) t += buf[w];
        buf[0] = t;
    }
    __syncthreads();
    return buf[0];
}

__device__ float block_max(float v, float* buf, int tid) {
    __syncthreads();
    v = wave_max(v);
    if ((tid & 31) == 0) buf[tid >> 5] = v;
    __syncthreads();
    if (tid == 0) {
        float t = buf[0];
        #pragma unroll
        for (int w = 1; w < FT_THREADS / 32; ++w) t = fmaxf(t, buf[w]);
        buf[0] = t;
    }
    __syncthreads();
    return buf[0];
}

__device__ __forceinline__ float sgnf(float v) {
    return (v > 0.f) ? 1.f : ((v < 0.f) ? -1.f : 0.f);
}

// One block per row: stage row in LDS, one-pass moments, bitonic sort for percentiles.
__global__ __launch_bounds__(FT_THREADS) void features_kernel(
    const float* __restrict__ sig, float* __restrict__ feats /* [B,32] */) {
    extern __shared__ float s[];          // L_SIG floats (64 KB)
    __shared__ float redbuf[FT_THREADS / 32];

    const int row = blockIdx.x;
    const int tid = threadIdx.x;
    const float* x = sig + (size_t)row * L_SIG;

    // coalesced load into LDS
    const float4* x4 = (const float4*)x;
    float4* s4 = (float4*)s;
    for (int i = tid; i < L_SIG / 4; i += FT_THREADS) s4[i] = x4[i];
    __syncthreads();

    // one-pass partials
    float s1 = 0.f, s2 = 0.f, s3 = 0.f, s4sum = 0.f;
    float sa = 0.f, sal = 0.f, pk = 0.f, pki = 0.f, zc = 0.f, mx = 0.f;
    for (int i = tid; i < L_SIG; i += FT_THREADS) {
        float v = s[i];
        float v2 = v * v;
        s1 += v; s2 += v2; s3 += v2 * v; s4sum += v2 * v2;
        float a = fabsf(v);
        sa += a;
        if (a > 0.f) sal += a * logf(a);
        mx = fmaxf(mx, a);
        if (i >= 1 && i <= L_SIG - 2) {
            float lft = s[i - 1], rgt = s[i + 1];
            if (v > lft && v > rgt) { pk += 1.f; pki += (float)i; }
        }
        if (i < L_SIG - 1) {
            if (sgnf(s[i + 1]) != sgnf(v)) zc += 1.f;
        }
    }
    s1 = block_sum(s1, redbuf, tid);
    s2 = block_sum(s2, redbuf, tid);
    s3 = block_sum(s3, redbuf, tid);
    s4sum = block_sum(s4sum, redbuf, tid);
    sa = block_sum(sa, redbuf, tid);
    sal = block_sum(sal, redbuf, tid);
    pk = block_sum(pk, redbuf, tid);
    pki = block_sum(pki, redbuf, tid);
    zc = block_sum(zc, redbuf, tid);
    mx = block_max(mx, redbuf, tid);

    const float invL = 1.f / (float)L_SIG;
    float mean = s1 * invL;
    float E2 = s2 * invL, E3 = s3 * invL, E4 = s4sum * invL;
    float var = E2 - mean * mean;
    float m2 = var;
    float m3 = E3 - 3.f * mean * E2 + 2.f * mean * mean * mean;
    float m4 = E4 - 4.f * mean * E3 + 6.f * mean * mean * E2 - 3.f * mean * mean * mean * mean;
    float stdv = sqrtf(fmaxf(var, 0.f));
    float sk = m3 / powf(m2, 1.5f);
    float ku = m4 / (m2 * m2) - 3.f;
    float ent = logf(sa) - sal / sa;         // -sum p*log p, p = |x|/S
    float rms = sqrtf(E2);
    float energy = s2;
    float peak = mx;
    float crest = (rms > 0.f) ? peak / rms : 0.f;
    float n_peaks = pk;
    float mean_peak = (pk > 0.f) ? pki / pk : 0.f;

    // bitonic sort of the row in LDS (ascending)
    for (unsigned k = 2; k <= (unsigned)L_SIG; k <<= 1) {
        for (unsigned j = k >> 1; j > 0; j >>= 1) {
            for (unsigned t = tid; t < (unsigned)L_SIG; t += FT_THREADS) {
                unsigned p = t ^ j;
                if (p > t) {
                    float a = s[t], b = s[p];
                    bool up = ((t & k) == 0);
                    if ((a > b) == up) { s[t] = b; s[p] = a; }
                }
            }
            __syncthreads();
        }
    }

    if (tid == 0) {
        float* f = feats + (size_t)row * 32;
        const float qs[5] = {10.f, 25.f, 50.f, 75.f, 90.f};
        float percs[5];
        #pragma unroll
        for (int q = 0; q < 5; ++q) {
            float pos = qs[q] * 0.01f * (float)(L_SIG - 1);
            int lo = (int)floorf(pos);
            if (lo > L_SIG - 2) lo = L_SIG - 2;
            float fr = pos - (float)lo;
            percs[q] = s[lo] + (s[lo + 1] - s[lo]) * fr;
        }
        f[0] = mean;  f[1] = stdv;  f[2] = var;  f[3] = sk;  f[4] = ku;  f[5] = ent;
        f[6] = percs[0]; f[7] = percs[1]; f[8] = percs[2]; f[9] = percs[3]; f[10] = percs[4];
        f[11] = n_peaks; f[12] = mean_peak; f[13] = rms; f[14] = zc;
        f[15] = energy;  f[16] = crest;
        #pragma unroll
        for (int i = 17; i < 25; ++i) f[i] = 0.05f;
        #pragma unroll
        for (int i = 25; i < 32; ++i) f[i] = 0.f;
    }
}

// One wave (32 lanes) per 16-row tile. f32 WMMA 16x16x4:
//  A (16x4):  lane l -> M = l&15, holds K = k0+koff, k0+koff+1 (koff = (l>=16)*2)
//  B (4x16):  lane l -> N = l&15, holds rows K = k0+koff, k0+koff+1
//  C/D 16x16: vgpr r -> M = r + 8*(l>=16), N = l&15
// All operands staged in LDS with uniform control flow -> no exec-mask churn
// around the v_wmma stream.
__global__ __launch_bounds__(32) void mlp_kernel(
    const float* __restrict__ feats /* [B,32] */,
    const float* __restrict__ W1 /* [25,64] */, const float* __restrict__ b1 /* [64] */,
    const float* __restrict__ W2 /* [64,32] */, const float* __restrict__ b2 /* [32] */,
    float* __restrict__ out /* [B,32] */) {
    __shared__ float w1s[32 * 64];   // zero-padded K: rows 25..31 = 0
    __shared__ float w2s[64 * 32];
    __shared__ float b1s[64];
    __shared__ float b2s[32];
    __shared__ float ats[16 * 32];   // feats tile
    __shared__ float hl[16 * 64];    // hidden activations

    const int lane = threadIdx.x;
    const int ml = lane & 15;
    const int koff = (lane >> 4) * 2;
    const int mrow = (lane >> 4) * 8;
    const int rowBase = blockIdx.x * 16;

    // ---- stage operands (uniform branches only) ----
    #pragma unroll
    for (int it = 0; it < (32 * 64) / 32; ++it) {
        const int i = it * 32 + lane;
        const int k = i >> 6;                    // wave-uniform per iteration
        w1s[i] = (k < 25) ? W1[i] : 0.f;
    }
    #pragma unroll
    for (int it = 0; it < (64 * 32) / 32; ++it) {
        const int i = it * 32 + lane;
        w2s[i] = W2[i];
    }
    #pragma unroll
    for (int it = 0; it < (16 * 32) / 32; ++it) {
        const int i = it * 32 + lane;
        ats[i] = feats[(size_t)rowBase * 32 + i];
    }
    b1s[lane] = b1[lane];
    b1s[lane + 32] = b1[lane + 32];
    b2s[lane] = b2[lane];
    __syncthreads();

    // ---- GEMM1: [16x32] @ [32x64] ----
    v8f acc[4] = {};
    #pragma unroll
    for (int ks = 0; ks < 8; ++ks) {
        const int k0 = ks * 4 + koff;
        v2f a;
        a.x = ats[ml * 32 + k0];
        a.y = ats[ml * 32 + k0 + 1];
        #pragma unroll
        for (int nt = 0; nt < 4; ++nt) {
            const int n = nt * 16 + ml;
            v2f b;
            b.x = w1s[k0 * 64 + n];
            b.y = w1s[(k0 + 1) * 64 + n];
            acc[nt] = __builtin_amdgcn_wmma_f32_16x16x4_f32(
                false, a, false, b, (short)0, acc[nt], false, false);
        }
    }

    // bias + ReLU, restage H (C-layout -> LDS row-major) for A-layout reload
    #pragma unroll
    for (int nt = 0; nt < 4; ++nt) {
        #pragma unroll
        for (int r = 0; r < 8; ++r) {
            const int m = r + mrow;
            const int n = nt * 16 + ml;
            float h = acc[nt][r] + b1s[n];
            hl[m * 64 + n] = fmaxf(h, 0.f);
        }
    }
    __syncthreads();

    // ---- GEMM2: [16x64] @ [64x32] ----
    v8f acc2[2] = {};
    #pragma unroll
    for (int ks = 0; ks < 16; ++ks) {
        const int k0 = ks * 4 + koff;
        v2f a;
        a.x = hl[ml * 64 + k0];
        a.y = hl[ml * 64 + k0 + 1];
        #pragma unroll
        for (int nt = 0; nt < 2; ++nt) {
            const int n = nt * 16 + ml;
            v2f b;
            b.x = w2s[k0 * 32 + n];
            b.y = w2s[(k0 + 1) * 32 + n];
            acc2[nt] = __builtin_amdgcn_wmma_f32_16x16x4_f32(
                false, a, false, b, (short)0, acc2[nt], false, false);
        }
    }

    #pragma unroll
    for (int nt = 0; nt < 2; ++nt) {
        #pragma unroll
        for (int r = 0; r < 8; ++r) {
            const int m = r + mrow;
            const int n = nt * 16 + ml;
            out[(size_t)(rowBase + m) * 32 + n] = acc2[nt][r] + b2s[n];
        }
    }
}

extern "C" void kernel_launch(void* const* d_in, const int* in_sizes, int n_in,
                              void* d_out, int out_size, void* d_ws, size_t ws_size,
                              hipStream_t stream) {
    const float* signals = (const float*)d_in[0];
    const float* W1 = (const float*)d_in[1];
    const float* b1 = (const float*)d_in[2];
    const float* W2 = (const float*)d_in[3];
    const float* b2 = (const float*)d_in[4];
    float* out = (float*)d_out;
    float* feats = (float*)d_ws;   // [4096, 32] padded feature matrix (512 KB)

    features_kernel<<<B_ROWS, FT_THREADS, L_SIG * sizeof(float), stream>>>(signals, feats);
    mlp_kernel<<<B_ROWS / 16, 32, 0, stream>>>(feats, W1, b1, W2, b2, out);
}